// Render_35287451304163
// MI455X (gfx1250) — compile-verified
//
#include <hip/hip_runtime.h>
#include <hip/hip_bf16.h>
#include <stdint.h>

// ---------------------------------------------------------------------------
// Gaussian splat renderer for gfx1250 (MI455X).
//  - maha quadratic form evaluated on the matrix pipe: 2x v_wmma_f32_16x16x4_f32
//    per 16-pixel x 16-gaussian tile (K=6 features padded to 8).
//  - compositing split low/high half-wave + shfl_xor(16) semiring combine.
//  - sorted gaussian records staged to LDS with global_load_async_to_lds_b128
//    (ASYNCcnt) double-buffered, shared by all 8 waves of the block.
// ---------------------------------------------------------------------------

typedef __attribute__((ext_vector_type(2))) float v2f;
typedef __attribute__((ext_vector_type(8))) float v8f;

#define NPTS     4096
#define IMG_W    256
#define IMG_H    256
#define TILE_SZ  64
#define GSTRIDE  16            // floats per gaussian record (64 B)
#define CHUNK    128           // gaussians per LDS stage (8 KB)
#define NCHUNK   (NPTS / CHUNK)
#define ALPHA_TH (1.0f / 255.0f)

// record layout (16 floats):
//  [0..5]  quadratic-form features gf0..gf5   (WMMA A, K=0..5)
//  [6..7]  0.0f pad                            (WMMA A, K=6..7)
//  [8]     opacity  [9..11] rgb  [12] z_raw
//  [13]    16-bit tile-overlap mask (bit t => overlaps 64x64 tile t), as uint
//  [14..15] pad

// ---------------------------------------------------------------------------
// Kernel 1: per-gaussian projection, 2D covariance, bbox, tile mask, features
// ---------------------------------------------------------------------------
__global__ __launch_bounds__(256) void k_preprocess(
    const float* __restrict__ xyz,   const float* __restrict__ covw,
    const float* __restrict__ colors,const float* __restrict__ opacity,
    const float* __restrict__ w2c,   const float* __restrict__ focal,
    const float* __restrict__ pp,
    float* __restrict__ rec, float* __restrict__ zbuf)
{
    int i = blockIdx.x * 256 + threadIdx.x;
    if (i >= NPTS) return;

    float Rm[3][3] = {{w2c[0], w2c[1], w2c[2]},
                      {w2c[4], w2c[5], w2c[6]},
                      {w2c[8], w2c[9], w2c[10]}};
    float tv[3]    = { w2c[3], w2c[7], w2c[11] };
    float fx = focal[0], fy = focal[1], px = pp[0], py = pp[1];

    float X = xyz[i*3+0], Y = xyz[i*3+1], Z = xyz[i*3+2];
    float camx = Rm[0][0]*X + Rm[0][1]*Y + Rm[0][2]*Z + tv[0];
    float camy = Rm[1][0]*X + Rm[1][1]*Y + Rm[1][2]*Z + tv[1];
    float camz = Rm[2][0]*X + Rm[2][1]*Y + Rm[2][2]*Z + tv[2];
    float zraw = camz;
    float z    = fmaxf(zraw, 1e-6f);
    float mx   = fx * camx / z + px;
    float my   = fy * camy / z + py;

    float Cw[3][3];
    #pragma unroll
    for (int r = 0; r < 3; ++r)
        #pragma unroll
        for (int c = 0; c < 3; ++c) Cw[r][c] = covw[i*9 + r*3 + c];

    // cov_cam = R * C * R^T
    float M[3][3], CC[3][3];
    #pragma unroll
    for (int r = 0; r < 3; ++r)
        #pragma unroll
        for (int c = 0; c < 3; ++c)
            M[r][c] = Rm[r][0]*Cw[0][c] + Rm[r][1]*Cw[1][c] + Rm[r][2]*Cw[2][c];
    #pragma unroll
    for (int r = 0; r < 3; ++r)
        #pragma unroll
        for (int c = 0; c < 3; ++c)
            CC[r][c] = M[r][0]*Rm[c][0] + M[r][1]*Rm[c][1] + M[r][2]*Rm[c][2];

    // J = [[fx/z, 0, -fx*camx/z^2], [0, fy/z, -fy*camy/z^2]]
    float j00 = fx / z, j02 = -fx * camx / (z*z);
    float j11 = fy / z, j12 = -fy * camy / (z*z);
    float JC0[3], JC1[3];
    #pragma unroll
    for (int c = 0; c < 3; ++c) {
        JC0[c] = j00*CC[0][c] + j02*CC[2][c];
        JC1[c] = j11*CC[1][c] + j12*CC[2][c];
    }
    float a  = JC0[0]*j00 + JC0[2]*j02 + 0.3f;   // cov2d[0][0]
    float b  = JC0[1]*j11 + JC0[2]*j12;          // cov2d[0][1]
    float c2 = JC1[0]*j00 + JC1[2]*j02;          // cov2d[1][0]
    float dd = JC1[1]*j11 + JC1[2]*j12 + 0.3f;   // cov2d[1][1]

    float disc   = sqrtf(fmaxf(0.25f*(a-dd)*(a-dd) + b*c2, 0.0f));
    float maxeig = fmaxf(0.5f*(a+dd) + disc, 1e-8f);
    float radius = fminf(3.0f * sqrtf(maxeig), 64.0f);
    float det    = a*dd - b*c2;
    float ic00   = dd / det;
    float ic11   = a  / det;
    float icsum  = -(b + c2) / det;

    float op = opacity[i];
    bool valid = (zraw > 0.1f) && (op > 1e-4f) && (radius > 0.25f) &&
                 (mx + radius >= -1.0f) && (mx - radius < (float)IMG_W + 1.0f) &&
                 (my + radius >= -1.0f) && (my - radius < (float)IMG_H + 1.0f);

    float radii = ceilf(radius);
    float rminx = fminf(fmaxf(floorf(mx - radii), 0.0f), (float)(IMG_W-1));
    float rmaxx = fminf(fmaxf(ceilf (mx + radii), 0.0f), (float)(IMG_W-1));
    float rminy = fminf(fmaxf(floorf(my - radii), 0.0f), (float)(IMG_H-1));
    float rmaxy = fminf(fmaxf(ceilf (my + radii), 0.0f), (float)(IMG_H-1));

    unsigned mask = 0u;
    if (valid) {
        #pragma unroll
        for (int t = 0; t < 16; ++t) {
            float x0 = (float)((t & 3)  * TILE_SZ);
            float y0 = (float)((t >> 2) * TILE_SZ);
            bool ov = (rmaxx >= x0) && (rminx < x0 + (float)TILE_SZ) &&
                      (rmaxy >= y0) && (rminy < y0 + (float)TILE_SZ);
            mask |= (ov ? 1u : 0u) << t;
        }
    }

    float gf3 = -(2.0f*ic00*mx + icsum*my);
    float gf4 = -(2.0f*ic11*my + icsum*mx);
    float gf5 = ic00*mx*mx + ic11*my*my + icsum*mx*my;

    float* o = rec + (size_t)i * GSTRIDE;
    o[0]  = ic00;  o[1]  = ic11;  o[2]  = icsum; o[3]  = gf3;
    o[4]  = gf4;   o[5]  = gf5;   o[6]  = 0.0f;  o[7]  = 0.0f;
    o[8]  = op;    o[9]  = colors[i*3+0]; o[10] = colors[i*3+1]; o[11] = colors[i*3+2];
    o[12] = zraw;  o[13] = __uint_as_float(mask); o[14] = 0.0f; o[15] = 0.0f;
    zbuf[i] = zraw;
}

// ---------------------------------------------------------------------------
// Kernel 2: stable O(N^2) rank-by-counting depth sort + record scatter
// ---------------------------------------------------------------------------
__global__ __launch_bounds__(256) void k_rank(
    const float* __restrict__ rec, const float* __restrict__ zbuf,
    float* __restrict__ srec)
{
    __shared__ float zs[NPTS];
    for (int j = threadIdx.x; j < NPTS; j += 256) zs[j] = zbuf[j];
    __syncthreads();

    int i  = blockIdx.x * 256 + threadIdx.x;
    float zi = zs[i];
    int rank = 0;
    for (int j = 0; j < NPTS; ++j) {
        float zj = zs[j];
        rank += ((zj < zi) || ((zj == zi) && (j < i))) ? 1 : 0;
    }
    const float4* s = (const float4*)(rec  + (size_t)i    * GSTRIDE);
    float4*       d = (float4*)      (srec + (size_t)rank * GSTRIDE);
    d[0] = s[0]; d[1] = s[1]; d[2] = s[2]; d[3] = s[3];
}

// ---------------------------------------------------------------------------
// Async global->LDS 16-byte copy (CDNA5 ASYNCcnt path).
// Low 32 bits of a generic shared pointer == LDS byte offset (aperture form).
// ---------------------------------------------------------------------------
__device__ __forceinline__ void async_copy16(const float* src, const float* dst_lds)
{
    uint32_t ldsoff = (uint32_t)(uintptr_t)dst_lds;
    asm volatile("global_load_async_to_lds_b128 %0, %1, off"
                 :: "v"(ldsoff), "v"(src) : "memory");
}

// ---------------------------------------------------------------------------
// Kernel 3: tiled front-to-back compositor.
//  block = 256 thr = 8 waves; wave owns 16 pixels (lanes 0-15 hold state).
//  Per 16-gaussian step: 2x v_wmma_f32_16x16x4_f32 -> maha[16g x 16px],
//  low half composites g0..7, high half g8..15, shfl_xor(16) combine.
// ---------------------------------------------------------------------------
__global__ __launch_bounds__(256) void k_render(
    const float* __restrict__ gsorted, const float* __restrict__ bg,
    float* __restrict__ out)
{
    __shared__ float sg[2][CHUNK * GSTRIDE];     // 2 x 8 KB double buffer

    const int tid  = threadIdx.x;
    const int lane = tid & 31;
    const int wave = tid >> 5;
    const int n    = lane & 15;                  // pixel column within wave
    const int h    = lane >> 4;                  // half-wave id

    const int tile = blockIdx.x >> 5;            // 16 tiles x 32 sub-blocks
    const int sub  = blockIdx.x & 31;
    const int tx   = tile & 3, ty = tile >> 2;

    const int plocal = wave * 16 + n;            // 0..127 within block
    const int gx = tx * TILE_SZ + (plocal & 63);
    const int gy = ty * TILE_SZ + sub * 2 + (plocal >> 6);
    const float cx = (float)gx + 0.5f;
    const float cy = (float)gy + 0.5f;

    // B matrix (pixel features, K x 16): vgpr v, lane l -> K = v + 2*(l>>4)
    // Pfeat = [cx^2, cy^2, cx*cy, cx, cy, 1, 0, 0]
    v2f b1, b2;
    b1[0] = h ? cx * cy : cx * cx;   // K2 : K0
    b1[1] = h ? cx      : cy * cy;   // K3 : K1
    b2[0] = h ? 0.0f    : cy;        // K6 : K4
    b2[1] = h ? 0.0f    : 1.0f;      // K7 : K5

    float PT = 1.0f, Pr = 0.f, Pg = 0.f, Pb = 0.f, Pa = 0.f, Pd = 0.f;

    auto issue = [&](int chunk, int buf) {
        const float* src = gsorted + (size_t)chunk * (CHUNK * GSTRIDE) + tid * 4;
        const float* dst = &sg[buf][tid * 4];
        async_copy16(src, dst);
        async_copy16(src + 1024, dst + 1024);    // second half of 8 KB chunk
    };

    issue(0, 0);
    for (int c = 0; c < NCHUNK; ++c) {
        const int buf = c & 1;
        if (c + 1 < NCHUNK) {
            issue(c + 1, buf ^ 1);
            asm volatile("s_wait_asynccnt 0x2" ::: "memory"); // chunk c landed
        } else {
            asm volatile("s_wait_asynccnt 0x0" ::: "memory");
        }
        __syncthreads();

        const float* base = &sg[buf][0];
        for (int s = 0; s < CHUNK / 16; ++s) {
            // A matrix (gaussian features, 16 x K): lane l row = l&15,
            // vgpr v -> K = v + 2*(l>>4)
            const float* grow = base + (s * 16 + n) * GSTRIDE;
            v2f a1, a2;
            a1[0] = grow[2*h + 0]; a1[1] = grow[2*h + 1];
            a2[0] = grow[4 + 2*h]; a2[1] = grow[5 + 2*h];

            v8f dmat = {0.f,0.f,0.f,0.f,0.f,0.f,0.f,0.f};
            dmat = __builtin_amdgcn_wmma_f32_16x16x4_f32(
                       false, a1, false, b1, (short)0, dmat, false, false);
            dmat = __builtin_amdgcn_wmma_f32_16x16x4_f32(
                       false, a2, false, b2, (short)0, dmat, false, false);

            // local composite: lane handles gaussians (8*h + v), v = 0..7
            float Tl = 1.0f, sr = 0.f, sgc = 0.f, sbc = 0.f, sa = 0.f, sd = 0.f;
            #pragma unroll
            for (int v = 0; v < 8; ++v) {
                const float* gq = base + (s * 16 + 8*h + v) * GSTRIDE;
                float maha = dmat[v];
                float opq = gq[8], crq = gq[9], cgq = gq[10], cbq = gq[11];
                float zq  = gq[12];
                unsigned msk = __float_as_uint(gq[13]);
                float gw = (maha <= 9.0f) ? __expf(-0.5f * maha) : 0.0f;
                float al = fminf(gw * opq, 0.99f);
                al = (al >= ALPHA_TH) ? al : 0.0f;
                al = ((msk >> tile) & 1u) ? al : 0.0f;
                float w = Tl * al;
                sr += w * crq; sgc += w * cgq; sbc += w * cbq;
                sa += w;       sd  += w * zq;
                Tl *= (1.0f - al);
            }

            // semiring combine: step = low ∘ high (depth order)
            float Th  = __shfl_xor(Tl, 16);
            float sr2 = sr  + Tl * __shfl_xor(sr, 16);
            float sg2 = sgc + Tl * __shfl_xor(sgc, 16);
            float sb2 = sbc + Tl * __shfl_xor(sbc, 16);
            float sa2 = sa  + Tl * __shfl_xor(sa, 16);
            float sd2 = sd  + Tl * __shfl_xor(sd, 16);

            Pr += PT * sr2; Pg += PT * sg2; Pb += PT * sb2;
            Pa += PT * sa2; Pd += PT * sd2;
            PT *= Tl * Th;
        }
        __syncthreads();
    }

    if (lane < 16) {
        float b0 = bg[0], b1v = bg[1], b2v = bg[2];
        float acc = Pa;
        float om  = 1.0f - acc;
        float r  = fminf(fmaxf(Pr + om * b0,  0.0f), 1.0f);
        float g  = fminf(fmaxf(Pg + om * b1v, 0.0f), 1.0f);
        float bb = fminf(fmaxf(Pb + om * b2v, 0.0f), 1.0f);
        float aw = fminf(fmaxf(acc, 0.0f), 1.0f);
        float dep = (acc > 0.0f) ? (Pd / fmaxf(acc, 1e-8f)) : 0.0f;
        float* o = out + ((size_t)gy * IMG_W + gx) * 5;
        o[0] = r; o[1] = g; o[2] = bb; o[3] = aw; o[4] = dep;
    }
}

// ---------------------------------------------------------------------------
extern "C" void kernel_launch(void* const* d_in, const int* in_sizes, int n_in,
                              void* d_out, int out_size, void* d_ws, size_t ws_size,
                              hipStream_t stream)
{
    const float* xyz     = (const float*)d_in[0];
    const float* covw    = (const float*)d_in[1];
    const float* colors  = (const float*)d_in[2];
    const float* opacity = (const float*)d_in[3];
    const float* w2c     = (const float*)d_in[4];
    const float* focal   = (const float*)d_in[5];
    const float* pp      = (const float*)d_in[6];
    const float* bg      = (const float*)d_in[7];
    float* out = (float*)d_out;

    float* rec   = (float*)d_ws;                 // 4096 * 16 floats (unsorted)
    float* zbuf  = rec  + (size_t)NPTS * GSTRIDE;// 4096 floats
    float* srec  = zbuf + NPTS;                  // 4096 * 16 floats (sorted)

    k_preprocess<<<NPTS/256, 256, 0, stream>>>(xyz, covw, colors, opacity,
                                               w2c, focal, pp, rec, zbuf);
    k_rank<<<NPTS/256, 256, 0, stream>>>(rec, zbuf, srec);

    const int nblocks = (IMG_W / TILE_SZ) * (IMG_H / TILE_SZ) * 32; // 512
    k_render<<<nblocks, 256, 0, stream>>>(srec, bg, out);
}